// GroupedGRU_80702435492217
// MI455X (gfx1250) — compile-verified
//
#include <hip/hip_runtime.h>
#include <math.h>

// ---------------------------------------------------------------------------
// Grouped GRU for MI455X (gfx1250, wave32).
//   Phase 1: input projection  gi[t][g][b][3H] = x_g[b,t,:] @ w_ih_g^T + b_ih
//            (parallel GEMM, V_WMMA_F32_16X16X4_F32, f32 precision)
//   Phase 2: persistent recurrence, one workgroup per group (one WGP each),
//            w_hh + h resident in LDS (320KB/WGP on CDNA5).
//            Gate-aligned N-tile ownership {nt, nt+8, nt+16} makes the GRU
//            gating wave-local (no LDS staging of gh), and gi[t] is
//            register-prefetched before the GEMM so HBM latency is hidden
//            behind the WMMA work on the serial path.
// Workspace requirement: T*G*B*3H floats = 188 MB in d_ws.
// ---------------------------------------------------------------------------

typedef __attribute__((ext_vector_type(2))) float v2f;
typedef __attribute__((ext_vector_type(8))) float v8f;

#define Gn 4
#define Bn 32
#define Tn 1000
#define Hn 128
#define In 128
#define H3 384

// LDS row strides (floats). 132 mod 64 == 4 banks -> the two 16-lane half
// groups of a wave hit disjoint bank sets for the float2 fragment loads.
#define WSTRIDE 132
#define HSTRIDE 132

// ---------------------------------------------------------------------------
// Phase 1: gi[((t*G+g)*B + b)*3H + k] = sum_i x[b,t,g*I+i] * w_ih[g,k,i] + b_ih
// grid = (G, T), block = 512 threads (16 waves). Wave w: Mtile = w&1,
// Ntiles = 3*(w>>1) .. +2. K swept 4 at a time with f32 WMMA.
// ---------------------------------------------------------------------------
__global__ __launch_bounds__(512) void gru_xproj(const float* __restrict__ x,
                                                 const float* __restrict__ w_ih,
                                                 const float* __restrict__ b_ih,
                                                 float* __restrict__ gi) {
  __shared__ float xs[Bn * WSTRIDE];  // 16.9 KB staged x slab [32 x 128]

  const int g   = blockIdx.x;
  const int t   = blockIdx.y;
  const int tid = threadIdx.x;

  // Stage x[b, t, g*128 + i] -> xs[b*WSTRIDE + i], coalesced (consecutive i).
  for (int r = 0; r < 8; ++r) {
    int idx = tid + 512 * r;  // 0..4095
    int b = idx >> 7;
    int i = idx & 127;
    xs[b * WSTRIDE + i] = x[((size_t)b * Tn + t) * (Gn * In) + (size_t)g * In + i];
  }
  __syncthreads();

  const int lane = tid & 31;
  const int wave = tid >> 5;   // 0..15
  const int lmod = lane & 15;
  const int lhi  = lane >> 4;  // 0/1 selects K pair (f32 WMMA A/B layout)
  const int m    = wave & 1;   // M tile (batch rows 0-15 / 16-31)
  const int ngrp = wave >> 1;  // 0..7 -> 3 N tiles each (24 total = 3H/16)
  const int row  = m * 16 + lmod;
  const int n0   = (ngrp * 3 + 0) * 16 + lmod;
  const int n1   = (ngrp * 3 + 1) * 16 + lmod;
  const int n2   = (ngrp * 3 + 2) * 16 + lmod;

  const float* wg = w_ih + (size_t)g * H3 * In;  // [3H][I], K-contiguous

  v8f acc0 = {}, acc1 = {}, acc2 = {};
  for (int k0 = 0; k0 < In; k0 += 4) {
    const int kk = k0 + lhi * 2;
    v2f a  = *(const v2f*)&xs[row * WSTRIDE + kk];
    v2f b0 = *(const v2f*)&wg[(size_t)n0 * In + kk];
    v2f b1 = *(const v2f*)&wg[(size_t)n1 * In + kk];
    v2f b2 = *(const v2f*)&wg[(size_t)n2 * In + kk];
    acc0 = __builtin_amdgcn_wmma_f32_16x16x4_f32(false, a, false, b0, (short)0, acc0, false, false);
    acc1 = __builtin_amdgcn_wmma_f32_16x16x4_f32(false, a, false, b1, (short)0, acc1, false, false);
    acc2 = __builtin_amdgcn_wmma_f32_16x16x4_f32(false, a, false, b2, (short)0, acc2, false, false);
  }

  // Fold b_ih here (parallel phase) so the serial phase skips it.
  const float bi0 = b_ih[g * H3 + n0];
  const float bi1 = b_ih[g * H3 + n1];
  const float bi2 = b_ih[g * H3 + n2];

  // D layout: element (M = v + 8*lhi (+16*m), N = col).
  float* base = gi + (size_t)(t * Gn + g) * Bn * H3;
#pragma unroll
  for (int v = 0; v < 8; ++v) {
    int b = m * 16 + lhi * 8 + v;
    base[(size_t)b * H3 + n0] = acc0[v] + bi0;
    base[(size_t)b * H3 + n1] = acc1[v] + bi1;
    base[(size_t)b * H3 + n2] = acc2[v] + bi2;
  }
}

// ---------------------------------------------------------------------------
// Phase 2: persistent recurrence. One block per group, 512 threads (16 waves),
// ~216 KB dynamic LDS (w_hh + h + b_hh) -> exactly one workgroup per WGP.
// Wave w owns Mtile (w&1) and N-tiles {ng, ng+8, ng+16} with ng = w>>1, so
// acc0/acc1/acc2 hold h_r/h_z/h_n for identical (b, j) elements and the GRU
// gating needs no cross-wave exchange.
// ---------------------------------------------------------------------------
__global__ __launch_bounds__(512) void gru_recur(const float* __restrict__ gi,
                                                 const float* __restrict__ w_hh,
                                                 const float* __restrict__ b_hh,
                                                 float* __restrict__ out) {
  extern __shared__ float smem[];
  float* whh  = smem;                  // [3H][WSTRIDE]  384*132 floats
  float* hbuf = whh + H3 * WSTRIDE;    // [B][HSTRIDE]   32*132 floats
  float* bhh  = hbuf + Bn * HSTRIDE;   // [3H]

  const int g   = blockIdx.x;
  const int tid = threadIdx.x;

  // Load this group's w_hh into LDS (coalesced), b_hh, zero h.
  for (int r = 0; r < (H3 * Hn) / 512; ++r) {
    int idx = tid + 512 * r;
    int n = idx >> 7;
    int k = idx & 127;
    whh[n * WSTRIDE + k] = w_hh[(size_t)g * H3 * Hn + idx];
  }
  if (tid < H3) bhh[tid] = b_hh[g * H3 + tid];
  for (int r = tid; r < Bn * HSTRIDE; r += 512) hbuf[r] = 0.0f;
  __syncthreads();

  const int lane = tid & 31;
  const int wave = tid >> 5;
  const int lmod = lane & 15;
  const int lhi  = lane >> 4;
  const int m    = wave & 1;   // Mtile: batch rows 0-15 / 16-31
  const int ng   = wave >> 1;  // 0..7: N-tiles {ng, ng+8, ng+16} (r/z/n gates)
  const int row  = m * 16 + lmod;
  const int jcol = ng * 16 + lmod;   // h-index column owned by this lane
  const int n0   = jcol;             // r-gate row of w_hh
  const int n1   = jcol + Hn;        // z-gate
  const int n2   = jcol + 2 * Hn;    // n-gate

  const float bz0 = 0.0f;  // (placeholder keeps biases symmetric if needed)
  (void)bz0;

  for (int t = 0; t < Tn; ++t) {
    // ---- register-prefetch gi[t] for this thread's 8 gating elements ----
    // Issued before the GEMM: LOADcnt traffic hides HBM latency behind WMMA.
    const float* gis = gi + (size_t)(t * Gn + g) * Bn * H3;
    float pir[8], piz[8], pin[8];
#pragma unroll
    for (int v = 0; v < 8; ++v) {
      int b = m * 16 + lhi * 8 + v;
      pir[v] = gis[(size_t)b * H3 + n0];
      piz[v] = gis[(size_t)b * H3 + n1];
      pin[v] = gis[(size_t)b * H3 + n2];
    }

    // ---- gh = h @ w_hh^T  (per wave: 1 Mtile x 3 gate-aligned N-tiles) ----
    v8f acc0 = {}, acc1 = {}, acc2 = {};
    for (int k0 = 0; k0 < Hn; k0 += 4) {
      const int kk = k0 + lhi * 2;
      v2f a  = *(const v2f*)&hbuf[row * HSTRIDE + kk];
      v2f b0 = *(const v2f*)&whh[n0 * WSTRIDE + kk];
      v2f b1 = *(const v2f*)&whh[n1 * WSTRIDE + kk];
      v2f b2 = *(const v2f*)&whh[n2 * WSTRIDE + kk];
      acc0 = __builtin_amdgcn_wmma_f32_16x16x4_f32(false, a, false, b0, (short)0, acc0, false, false);
      acc1 = __builtin_amdgcn_wmma_f32_16x16x4_f32(false, a, false, b1, (short)0, acc1, false, false);
      acc2 = __builtin_amdgcn_wmma_f32_16x16x4_f32(false, a, false, b2, (short)0, acc2, false, false);
    }

    __syncthreads();  // all GEMM reads of hbuf complete before h update

    // ---- wave-local gating: acc0/1/2[v] are h_r/h_z/h_n at (b, jcol) ----
    const float bh0 = bhh[n0];
    const float bh1 = bhh[n1];
    const float bh2 = bhh[n2];
#pragma unroll
    for (int v = 0; v < 8; ++v) {
      int b = m * 16 + lhi * 8 + v;
      float hr = acc0[v] + bh0;
      float hz = acc1[v] + bh1;
      float hn = acc2[v] + bh2;
      float rg = 1.0f / (1.0f + expf(-(pir[v] + hr)));
      float zg = 1.0f / (1.0f + expf(-(piz[v] + hz)));
      float ngt = tanhf(pin[v] + rg * hn);
      float hp = hbuf[b * HSTRIDE + jcol];
      float hv = (1.0f - zg) * ngt + zg * hp;  // in-place safe: only (b,jcol)
      hbuf[b * HSTRIDE + jcol] = hv;
      out[((size_t)b * Tn + t) * (Gn * Hn) + (size_t)g * Hn + jcol] = hv;
    }

    __syncthreads();  // h fully updated before next step's GEMM
  }
}

// ---------------------------------------------------------------------------
extern "C" void kernel_launch(void* const* d_in, const int* in_sizes, int n_in,
                              void* d_out, int out_size, void* d_ws, size_t ws_size,
                              hipStream_t stream) {
  const float* x    = (const float*)d_in[0];
  const float* w_ih = (const float*)d_in[1];
  const float* w_hh = (const float*)d_in[2];
  const float* b_ih = (const float*)d_in[3];
  const float* b_hh = (const float*)d_in[4];
  float* out = (float*)d_out;
  float* gi  = (float*)d_ws;  // needs T*G*B*3H*4 = 188 MB

  gru_xproj<<<dim3(Gn, Tn), dim3(512), 0, stream>>>(x, w_ih, b_ih, gi);

  size_t lds_bytes =
      (size_t)(H3 * WSTRIDE + Bn * HSTRIDE + H3) * sizeof(float);
  gru_recur<<<dim3(Gn), dim3(512), lds_bytes, stream>>>(gi, w_hh, b_hh, out);
}